// AccFlowSupervise_77309411328467
// MI455X (gfx1250) — compile-verified
//
#include <hip/hip_runtime.h>
#include <math.h>

typedef __attribute__((ext_vector_type(2))) float v2f;
typedef __attribute__((ext_vector_type(8))) float v8f;

#define KNN_EPS 1e-8f
#define BIGF 3.0e38f

// ---------------------------------------------------------------------------
// Tiny prep: pose_0to1 = inv(pose1) @ pose0 for SE(3):
//   R = R1^T R0 ;  t = R1^T (t0 - t1).   Writes 12 floats per batch to ws.
// ---------------------------------------------------------------------------
__global__ void pose_prep_kernel(const float* __restrict__ pose0,
                                 const float* __restrict__ pose1,
                                 float* __restrict__ rt, int B) {
  int b = blockIdx.x * blockDim.x + threadIdx.x;
  if (b >= B) return;
  const float* P0 = pose0 + (size_t)b * 16;
  const float* P1 = pose1 + (size_t)b * 16;
  float* o = rt + (size_t)b * 12;
  // R[i][j] = sum_k R1[k][i] * R0[k][j]
  #pragma unroll
  for (int i = 0; i < 3; ++i)
    #pragma unroll
    for (int j = 0; j < 3; ++j) {
      float s = 0.f;
      #pragma unroll
      for (int k = 0; k < 3; ++k) s += P1[k * 4 + i] * P0[k * 4 + j];
      o[i * 3 + j] = s;
    }
  float dt0 = P0[3] - P1[3], dt1 = P0[7] - P1[7], dt2 = P0[11] - P1[11];
  #pragma unroll
  for (int i = 0; i < 3; ++i)
    o[9 + i] = P1[0 * 4 + i] * dt0 + P1[1 * 4 + i] * dt1 + P1[2 * 4 + i] * dt2;
}

// ---------------------------------------------------------------------------
// lane <-> lane^16 exchange via ds_swizzle SWAPX16 (xor=0x10,or=0,and=0x1f)
// ---------------------------------------------------------------------------
__device__ __forceinline__ float swap16_f(float x) {
  return __int_as_float(__builtin_amdgcn_ds_swizzle(__float_as_int(x), 0x401F));
}
__device__ __forceinline__ int swap16_i(int x) {
  return __builtin_amdgcn_ds_swizzle(x, 0x401F);
}

// Branchless insert into sorted top-3 (v0 <= v1 <= v2)
__device__ __forceinline__ void insert3(float d, int j,
                                        float& v0, float& v1, float& v2,
                                        int& j0, int& j1, int& j2) {
  bool c2 = d < v2;
  bool c1 = d < v1;
  bool c0 = d < v0;
  v2 = c1 ? v1 : (c2 ? d : v2);
  j2 = c1 ? j1 : (c2 ? j : j2);
  v1 = c0 ? v0 : (c1 ? d : v1);
  j1 = c0 ? j0 : (c1 ? j : j1);
  v0 = c0 ? d : v0;
  j0 = c0 ? j : j0;
}

// ---------------------------------------------------------------------------
// One wave (32 lanes) handles 16 pc0 queries. Sweeps all M pc1 points in
// 16-row tiles with V_WMMA_F32_16X16X4_F32:
//   A (per-iter) : 16 pc1 rows  [rx, ry, rz, |r|^2]
//   B (invariant): 16 queries   [-2qx, -2qy, -2qz, 1]^T
//   D[j][i] = |r_j|^2 - 2 q_i . r_j      (add |q_i|^2 at the very end)
// Lane l owns query i = i0 + (l%16); its v8f D slice holds 8 candidate j's
// (j = jbase + k + 8*(l>=16)) -> lane-local top-3, merged across halves once.
// ---------------------------------------------------------------------------
__global__ __launch_bounds__(32) void knn_flow_kernel(
    const float* __restrict__ pc0, const float* __restrict__ pc1,
    const float* __restrict__ flow1, const float* __restrict__ rt,
    float* __restrict__ out, int B, int N, int M) {
  const int tilesPerBatch = N >> 4;
  const int b  = blockIdx.x / tilesPerBatch;
  const int i0 = (blockIdx.x % tilesPerBatch) << 4;
  const int lane = threadIdx.x;      // 0..31
  const int li   = lane & 15;
  const int hi   = lane >> 4;        // 0 or 1

  const float* R = rt + (size_t)b * 12;
  const int i = i0 + li;

  // transform query point
  const float px = pc0[((size_t)b * N + i) * 3 + 0];
  const float py = pc0[((size_t)b * N + i) * 3 + 1];
  const float pz = pc0[((size_t)b * N + i) * 3 + 2];
  const float qx = R[0] * px + R[1] * py + R[2] * pz + R[9];
  const float qy = R[3] * px + R[4] * py + R[5] * pz + R[10];
  const float qz = R[6] * px + R[7] * py + R[8] * pz + R[11];
  const float q2 = qx * qx + qy * qy + qz * qz;

  // output 2: pose_flow (written once, by low half)
  if (hi == 0) {
    const size_t o = (size_t)B * N * 3 + ((size_t)b * N + i) * 3;
    out[o + 0] = qx - px;
    out[o + 1] = qy - py;
    out[o + 2] = qz - pz;
  }

  // loop-invariant B fragment: col i = [-2qx, -2qy, -2qz, 1]
  v2f bfrag;
  bfrag.x = hi ? (-2.0f * qz) : (-2.0f * qx);
  bfrag.y = hi ? 1.0f : (-2.0f * qy);

  float v0 = BIGF, v1 = BIGF, v2 = BIGF;
  int   j0 = 0,    j1 = 0,    j2 = 0;

  const float* pc1b = pc1 + (size_t)b * M * 3;
  for (int jb = 0; jb < M; jb += 16) {
    const int jr = jb + li;  // A row handled by this lane (both halves: M=l%16)
    const float rx = pc1b[jr * 3 + 0];
    const float ry = pc1b[jr * 3 + 1];
    const float rz = pc1b[jr * 3 + 2];
    const float rr = rx * rx + ry * ry + rz * rz;
    v2f afrag;
    afrag.x = hi ? rz : rx;
    afrag.y = hi ? rr : ry;

    v8f c = {};
    v8f d = __builtin_amdgcn_wmma_f32_16x16x4_f32(
        /*neg_a=*/false, afrag, /*neg_b=*/false, bfrag,
        /*c_mod=*/(short)0, c, /*reuse_a=*/false, /*reuse_b=*/false);

    // fast wave-coherent skip: does any of my 8 candidates beat 3rd best?
    float m = fminf(fminf(fminf(d[0], d[1]), fminf(d[2], d[3])),
                    fminf(fminf(d[4], d[5]), fminf(d[6], d[7])));
    if (m < v2) {
      const int jlo = jb + (hi << 3);  // this lane's D rows: j = jlo + k
      insert3(d[0], jlo + 0, v0, v1, v2, j0, j1, j2);
      insert3(d[1], jlo + 1, v0, v1, v2, j0, j1, j2);
      insert3(d[2], jlo + 2, v0, v1, v2, j0, j1, j2);
      insert3(d[3], jlo + 3, v0, v1, v2, j0, j1, j2);
      insert3(d[4], jlo + 4, v0, v1, v2, j0, j1, j2);
      insert3(d[5], jlo + 5, v0, v1, v2, j0, j1, j2);
      insert3(d[6], jlo + 6, v0, v1, v2, j0, j1, j2);
      insert3(d[7], jlo + 7, v0, v1, v2, j0, j1, j2);
    }
  }

  // merge the two lane-halves (disjoint j sets for the same query i)
  {
    float ov0 = swap16_f(v0), ov1 = swap16_f(v1), ov2 = swap16_f(v2);
    int   oj0 = swap16_i(j0), oj1 = swap16_i(j1), oj2 = swap16_i(j2);
    insert3(ov0, oj0, v0, v1, v2, j0, j1, j2);
    insert3(ov1, oj1, v0, v1, v2, j0, j1, j2);
    insert3(ov2, oj2, v0, v1, v2, j0, j1, j2);
  }

  // output 1: flow_interp (low half writes)
  if (hi == 0) {
    const float d0 = sqrtf(fmaxf(v0 + q2, 0.0f));
    const float d1 = sqrtf(fmaxf(v1 + q2, 0.0f));
    const float d2 = sqrtf(fmaxf(v2 + q2, 0.0f));
    float w0 = 1.0f / (d0 + KNN_EPS);
    float w1 = 1.0f / (d1 + KNN_EPS);
    float w2 = 1.0f / (d2 + KNN_EPS);
    const float ws = 1.0f / (w0 + w1 + w2);
    w0 *= ws; w1 *= ws; w2 *= ws;
    const float* fb = flow1 + (size_t)b * M * 3;
    const size_t o = ((size_t)b * N + i) * 3;
    out[o + 0] = w0 * fb[j0 * 3 + 0] + w1 * fb[j1 * 3 + 0] + w2 * fb[j2 * 3 + 0];
    out[o + 1] = w0 * fb[j0 * 3 + 1] + w1 * fb[j1 * 3 + 1] + w2 * fb[j2 * 3 + 1];
    out[o + 2] = w0 * fb[j0 * 3 + 2] + w1 * fb[j1 * 3 + 2] + w2 * fb[j2 * 3 + 2];
  }
}

// ---------------------------------------------------------------------------
extern "C" void kernel_launch(void* const* d_in, const int* in_sizes, int n_in,
                              void* d_out, int out_size, void* d_ws, size_t ws_size,
                              hipStream_t stream) {
  const float* pc0   = (const float*)d_in[0];
  const float* pc1   = (const float*)d_in[1];
  const float* flow1 = (const float*)d_in[2];
  const float* pose0 = (const float*)d_in[3];
  const float* pose1 = (const float*)d_in[4];

  const int B = in_sizes[3] / 16;           // poses are [B,4,4]
  const int N = in_sizes[0] / (B * 3);      // pc0 [B,N,3]
  const int M = in_sizes[1] / (B * 3);      // pc1 [B,M,3]

  float* rt = (float*)d_ws;                 // 12 floats per batch

  pose_prep_kernel<<<1, B, 0, stream>>>(pose0, pose1, rt, B);

  const int blocks = B * (N / 16);          // one wave (32 threads) per 16 queries
  knn_flow_kernel<<<blocks, 32, 0, stream>>>(pc0, pc1, flow1, rt,
                                             (float*)d_out, B, N, M);
}